// GeneSetPoolingAggregator_72782515798445
// MI455X (gfx1250) — compile-verified
//
#include <hip/hip_runtime.h>
#include <stdint.h>

#define BATCH    8
#define NGENES   20000
#define NSEG     4096
#define FEATS    64
#define MAXCNT   96   // counts are in [10, 90)

#if defined(__HIP_DEVICE_COMPILE__) && defined(__has_builtin)
#  if __has_builtin(__builtin_amdgcn_global_load_async_to_lds_b32)
#    define HAVE_ASYNC_BUILTIN 1
#  endif
#  if __has_builtin(__builtin_amdgcn_s_wait_asynccnt)
#    define HAVE_WAIT_ASYNC_BUILTIN 1
#  endif
#endif

#if defined(__HIP_DEVICE_COMPILE__)
typedef __attribute__((address_space(1))) int glob_i32;
typedef __attribute__((address_space(3))) int lds_i32;
#endif

// ---------------------------------------------------------------------------
// Kernel 1: exclusive prefix-sum of counts[4096] -> offsets[4096]
// Single block, 1024 threads, 4 counts per thread, Hillis-Steele scan in LDS.
// ---------------------------------------------------------------------------
__global__ __launch_bounds__(1024)
void genesets_scan_kernel(const int* __restrict__ counts,
                          int* __restrict__ offsets) {
#if defined(__HIP_DEVICE_COMPILE__)
  __shared__ int s_part[1024];
  const int t = threadIdx.x;
  const int4 c = reinterpret_cast<const int4*>(counts)[t];
  const int local = c.x + c.y + c.z + c.w;
  s_part[t] = local;
  __syncthreads();
#pragma unroll
  for (int off = 1; off < 1024; off <<= 1) {
    const int v = (t >= off) ? s_part[t - off] : 0;
    __syncthreads();
    s_part[t] += v;
    __syncthreads();
  }
  const int base = s_part[t] - local;  // exclusive prefix of this 4-chunk
  offsets[4 * t + 0] = base;
  offsets[4 * t + 1] = base + c.x;
  offsets[4 * t + 2] = base + c.x + c.y;
  offsets[4 * t + 3] = base + c.x + c.y + c.z;
#endif
}

// ---------------------------------------------------------------------------
// Kernel 2: one block per geneset. 128 threads = 8 batches x 16 lanes,
// each thread accumulates a float4 (4 feats) over the segment's items.
// Indices staged once into LDS via CDNA5 async global->LDS copy.
// ---------------------------------------------------------------------------
__global__ __launch_bounds__(128)
void genesets_pool_kernel(const float* __restrict__ gene,
                          const int* __restrict__ flat_idx,
                          const int* __restrict__ counts,
                          const int* __restrict__ offsets,
                          float* __restrict__ out) {
#if defined(__HIP_DEVICE_COMPILE__)
  __shared__ int s_idx[MAXCNT];

  const int s   = blockIdx.x;
  const int tid = threadIdx.x;
  const int c   = counts[s];
  const int o   = offsets[s];

  // ---- stage this segment's gene indices into LDS (ASYNCcnt path) ----
  if (tid < c) {
    const int* gp = flat_idx + o + tid;
#if defined(HAVE_ASYNC_BUILTIN)
    __builtin_amdgcn_global_load_async_to_lds_b32(
        (glob_i32*)(uintptr_t)gp,
        (lds_i32*)(uint32_t)(uintptr_t)&s_idx[tid],
        0, 0);
#else
    asm volatile("global_load_async_to_lds_b32 %0, %1, off"
                 :
                 : "v"((uint32_t)(uintptr_t)&s_idx[tid]),
                   "v"((uint64_t)(uintptr_t)gp)
                 : "memory");
#endif
  }
#if defined(HAVE_WAIT_ASYNC_BUILTIN)
  __builtin_amdgcn_s_wait_asynccnt(0);
#else
  asm volatile("s_wait_asynccnt 0" ::: "memory");
#endif
  __syncthreads();

  // ---- accumulate: thread = (batch b, feat group fb..fb+3) ----
  const int fb = (tid & 15) << 2;  // 16 lanes x float4 = one 256B gene row
  const int b  = tid >> 4;         // batch 0..7
  const float* __restrict__ base = gene + (size_t)b * NGENES * FEATS + fb;

  float4 a0 = {0.f, 0.f, 0.f, 0.f};
  float4 a1 = a0, a2 = a0, a3 = a0;

  int i = 0;
  for (; i + 4 <= c; i += 4) {
    const int g0 = s_idx[i + 0];
    const int g1 = s_idx[i + 1];
    const int g2 = s_idx[i + 2];
    const int g3 = s_idx[i + 3];
    if (i + 8 < c)  // pre-touch a future gathered row (global_prefetch_b8)
      __builtin_prefetch(base + (size_t)s_idx[i + 8] * FEATS, 0, 0);
    const float4 v0 = *(const float4*)(base + (size_t)g0 * FEATS);
    const float4 v1 = *(const float4*)(base + (size_t)g1 * FEATS);
    const float4 v2 = *(const float4*)(base + (size_t)g2 * FEATS);
    const float4 v3 = *(const float4*)(base + (size_t)g3 * FEATS);
    a0.x += v0.x; a0.y += v0.y; a0.z += v0.z; a0.w += v0.w;
    a1.x += v1.x; a1.y += v1.y; a1.z += v1.z; a1.w += v1.w;
    a2.x += v2.x; a2.y += v2.y; a2.z += v2.z; a2.w += v2.w;
    a3.x += v3.x; a3.y += v3.y; a3.z += v3.z; a3.w += v3.w;
  }
  for (; i < c; ++i) {
    const float4 v = *(const float4*)(base + (size_t)s_idx[i] * FEATS);
    a0.x += v.x; a0.y += v.y; a0.z += v.z; a0.w += v.w;
  }

  const float inv = 1.0f / (float)c;
  float4 r;
  r.x = (a0.x + a1.x + a2.x + a3.x) * inv;
  r.y = (a0.y + a1.y + a2.y + a3.y) * inv;
  r.z = (a0.z + a1.z + a2.z + a3.z) * inv;
  r.w = (a0.w + a1.w + a2.w + a3.w) * inv;

  *(float4*)(out + ((size_t)b * NSEG + s) * FEATS + fb) = r;
#endif
}

// ---------------------------------------------------------------------------
// Host launcher. Inputs: gene_output(f32), flat_indices(i32),
// segment_ids(i32, unused: segments are contiguous), counts(i32).
// ---------------------------------------------------------------------------
extern "C" void kernel_launch(void* const* d_in, const int* in_sizes, int n_in,
                              void* d_out, int out_size, void* d_ws, size_t ws_size,
                              hipStream_t stream) {
  const float* gene     = (const float*)d_in[0];
  const int*   flat_idx = (const int*)d_in[1];
  const int*   counts   = (const int*)d_in[3];
  int* offsets = (int*)d_ws;  // 4096 ints of scratch

  genesets_scan_kernel<<<1, 1024, 0, stream>>>(counts, offsets);
  genesets_pool_kernel<<<NSEG, 128, 0, stream>>>(gene, flat_idx, counts,
                                                 offsets, (float*)d_out);
  (void)in_sizes; (void)n_in; (void)out_size; (void)ws_size;
}